// JumpingKnowledge_64656437674327
// MI455X (gfx1250) — compile-verified
//
#include <hip/hip_runtime.h>
#include <hip/hip_bf16.h>

// Problem constants (match reference)
#define NNODES   50000
#define NEDGES   600000
#define DFEAT    128
#define HDIM     128
#define NCLS     64
#define NLAYERS  3
#define XCDIM    512          // D + L*H
#define ROWTILES (NNODES/16)  // 3125 exact
#define LN_EPS   1e-5f

typedef __attribute__((ext_vector_type(2))) float v2f;
typedef __attribute__((ext_vector_type(8))) float v8f;

// ---------------------------------------------------------------------------
// Small utility kernels
// ---------------------------------------------------------------------------
__global__ void fill_ones_kernel(float* __restrict__ p, int n) {
  int i = blockIdx.x * blockDim.x + threadIdx.x;
  if (i < n) p[i] = 1.0f;
}

__global__ void deg_accum_kernel(const int* __restrict__ dst, float* __restrict__ deg, int e) {
  int i = blockIdx.x * blockDim.x + threadIdx.x;
  if (i < e) atomicAdd(&deg[dst[i]], 1.0f);
}

__global__ void rsqrt_inplace_kernel(float* __restrict__ p, int n) {
  int i = blockIdx.x * blockDim.x + threadIdx.x;
  if (i < n) p[i] = rsqrtf(p[i]);
}

// xc[:, 0:128] = x   (float4 granularity: N*32 threads)
__global__ void copy_x_kernel(const float* __restrict__ x, float* __restrict__ xc) {
  int i = blockIdx.x * blockDim.x + threadIdx.x;
  if (i < NNODES * (DFEAT / 4)) {
    int row = i >> 5, c4 = i & 31;
    ((float4*)(xc + (size_t)row * XCDIM))[c4] =
        ((const float4*)(x + (size_t)row * DFEAT))[c4];
  }
}

// agg = hW * dinv^2   (self-loop term, also zero-initializes agg for scatter)
__global__ void selfloop_kernel(const float* __restrict__ hW, const float* __restrict__ dinv,
                                float* __restrict__ agg) {
  int i = blockIdx.x * blockDim.x + threadIdx.x;
  if (i < NNODES * (HDIM / 4)) {
    int row = i >> 5, c4 = i & 31;
    float d2 = dinv[row]; d2 *= d2;
    float4 v = ((const float4*)(hW + (size_t)row * HDIM))[c4];
    v.x *= d2; v.y *= d2; v.z *= d2; v.w *= d2;
    ((float4*)(agg + (size_t)row * HDIM))[c4] = v;
  }
}

// ---------------------------------------------------------------------------
// Edge scatter: agg[dst] += hW[src] * dinv[src]*dinv[dst]
// 32 lanes per edge, 4 features per lane -> global_atomic_add_f32
// ---------------------------------------------------------------------------
__global__ void scatter_kernel(const int* __restrict__ src, const int* __restrict__ dst,
                               const float* __restrict__ dinv,
                               const float* __restrict__ hW, float* __restrict__ agg) {
  int tid = blockIdx.x * blockDim.x + threadIdx.x;
  int e = tid >> 5;
  if (e >= NEDGES) return;
  int f0 = (tid & 31) * 4;
  int s = src[e], d = dst[e];
  float coef = dinv[s] * dinv[d];
  float4 v = *(const float4*)(hW + (size_t)s * HDIM + f0);
  float* ap = agg + (size_t)d * HDIM + f0;
  atomicAdd(ap + 0, v.x * coef);
  atomicAdd(ap + 1, v.y * coef);
  atomicAdd(ap + 2, v.z * coef);
  atomicAdd(ap + 3, v.w * coef);
}

// ---------------------------------------------------------------------------
// Bias + LayerNorm + ReLU, one wave32 per node (4 feats/lane), writes into xc slice
// ---------------------------------------------------------------------------
__device__ __forceinline__ float wave_sum32(float v) {
#pragma unroll
  for (int d = 16; d >= 1; d >>= 1) v += __shfl_xor(v, d, 32);
  return v;
}

__global__ void post_ln_kernel(const float* __restrict__ agg, const float* __restrict__ bg,
                               const float* __restrict__ g, const float* __restrict__ b,
                               float* __restrict__ xc, int coloff) {
  int wid = (blockIdx.x * blockDim.x + threadIdx.x) >> 5;  // node index
  if (wid >= NNODES) return;
  int lane = threadIdx.x & 31;
  int f0 = lane * 4;
  float4 t  = *(const float4*)(agg + (size_t)wid * HDIM + f0);
  float4 bb = *(const float4*)(bg + f0);
  t.x += bb.x; t.y += bb.y; t.z += bb.z; t.w += bb.w;
  float mu = wave_sum32(t.x + t.y + t.z + t.w) * (1.0f / HDIM);
  float dx = t.x - mu, dy = t.y - mu, dz = t.z - mu, dw = t.w - mu;
  float var = wave_sum32(dx * dx + dy * dy + dz * dz + dw * dw) * (1.0f / HDIM);
  float inv = rsqrtf(var + LN_EPS);
  float4 gg = *(const float4*)(g + f0);
  float4 be = *(const float4*)(b + f0);
  float4 y;
  y.x = fmaxf(gg.x * dx * inv + be.x, 0.0f);
  y.y = fmaxf(gg.y * dy * inv + be.y, 0.0f);
  y.z = fmaxf(gg.z * dz * inv + be.z, 0.0f);
  y.w = fmaxf(gg.w * dw * inv + be.w, 0.0f);
  *(float4*)(xc + (size_t)wid * XCDIM + coloff + f0) = y;
}

// ---------------------------------------------------------------------------
// WMMA fp32 GEMM: C[16-row tile] = A @ B, via V_WMMA_F32_16X16X4_F32
// A-fragment layout (32-bit A 16x4): lanes 0-15 -> K=kk..kk+1, lanes 16-31 -> K=kk+2..kk+3
// B-fragment layout (4x16): same K split across lane halves, N = lane&15
// C/D layout: VGPR v, lane l -> row = v + 8*(l>=16), col = l&15
// ---------------------------------------------------------------------------

// GCN layer GEMM: [N,128] x [128,128] -> [N,128]. 8 waves/block, 1 row tile/block.
__global__ void gemm128_kernel(const float* __restrict__ A, int lda,
                               const float* __restrict__ B,   // [128,128] row-major
                               float* __restrict__ C) {       // [N,128]
  __shared__ float lds_a[16 * 132];  // +4 pad: 32 lanes -> 32 distinct banks
  const int row0 = blockIdx.x * 16;
  const int t = threadIdx.x;
  {  // stage A tile (coalesced: 16 threads share a row, 8 floats each)
    int r = t >> 4, c0 = (t & 15) * 8;
    const float4* sp = (const float4*)(A + (size_t)(row0 + r) * lda + c0);
    float4* dp = (float4*)(&lds_a[r * 132 + c0]);
    dp[0] = sp[0]; dp[1] = sp[1];
  }
  __syncthreads();
  const int wid = t >> 5;           // col tile 0..7
  const int lane = t & 31;
  const int lo = lane & 15;
  const int koff = (lane >> 4) << 1;  // 0 or 2
  const float* arow = &lds_a[lo * 132];
  const float* bcol = B + wid * 16 + lo;
  v8f acc = {};
#pragma unroll 4
  for (int kk = 0; kk < 128; kk += 4) {
    v2f a, b;
    a.x = arow[kk + koff];
    a.y = arow[kk + koff + 1];
    b.x = bcol[(size_t)(kk + koff) * 128];
    b.y = bcol[(size_t)(kk + koff + 1) * 128];
    acc = __builtin_amdgcn_wmma_f32_16x16x4_f32(false, a, false, b, (short)0, acc, false, false);
  }
  const int hi = lane >> 4;
  float* cp = C + (size_t)(row0 + hi * 8) * 128 + wid * 16 + lo;
#pragma unroll
  for (int v = 0; v < 8; ++v) cp[(size_t)v * 128] = acc[v];
}

// MLP layer 1: [N,512] x [512,128] + b1, ReLU -> z1 [N,128]
__global__ void mlp1_kernel(const float* __restrict__ xc, const float* __restrict__ W1,
                            const float* __restrict__ b1, float* __restrict__ z1) {
  __shared__ float lds_a[16 * 516];  // 33 KB, pad keeps banks conflict-free
  const int row0 = blockIdx.x * 16;
  const int t = threadIdx.x;
  {  // stage 16x512 xc tile: 32 floats (8x float4) per thread
    int r = t >> 4, c0 = (t & 15) * 32;
    const float4* sp = (const float4*)(xc + (size_t)(row0 + r) * XCDIM + c0);
    float4* dp = (float4*)(&lds_a[r * 516 + c0]);
#pragma unroll
    for (int j = 0; j < 8; ++j) dp[j] = sp[j];
  }
  __syncthreads();
  const int wid = t >> 5;
  const int lane = t & 31;
  const int lo = lane & 15;
  const int koff = (lane >> 4) << 1;
  const float* arow = &lds_a[lo * 516];
  const float* bcol = W1 + wid * 16 + lo;
  v8f acc = {};
#pragma unroll 4
  for (int kk = 0; kk < 512; kk += 4) {
    v2f a, b;
    a.x = arow[kk + koff];
    a.y = arow[kk + koff + 1];
    b.x = bcol[(size_t)(kk + koff) * 128];
    b.y = bcol[(size_t)(kk + koff + 1) * 128];
    acc = __builtin_amdgcn_wmma_f32_16x16x4_f32(false, a, false, b, (short)0, acc, false, false);
  }
  const int hi = lane >> 4;
  const float bias = b1[wid * 16 + lo];
  float* cp = z1 + (size_t)(row0 + hi * 8) * 128 + wid * 16 + lo;
#pragma unroll
  for (int v = 0; v < 8; ++v) cp[(size_t)v * 128] = fmaxf(acc[v] + bias, 0.0f);
}

// MLP layer 2 + softmax: one wave owns a 16x64 logit strip (4 WMMA col tiles,
// A-fragment reused across the 4 WMMAs), then 16-lane shfl_xor softmax.
__global__ void mlp2_softmax_kernel(const float* __restrict__ z1, const float* __restrict__ W2,
                                    const float* __restrict__ b2, float* __restrict__ out) {
  int gwid = (blockIdx.x * blockDim.x + threadIdx.x) >> 5;  // wave-uniform
  if (gwid >= ROWTILES) return;
  const int row0 = gwid * 16;
  const int lane = threadIdx.x & 31;
  const int lo = lane & 15;
  const int koff = (lane >> 4) << 1;
  const float* arow = z1 + (size_t)(row0 + lo) * 128;
  v8f acc[4];
#pragma unroll
  for (int tt = 0; tt < 4; ++tt) acc[tt] = (v8f){};
  for (int kk = 0; kk < 128; kk += 4) {
    v2f a;
    a.x = arow[kk + koff];
    a.y = arow[kk + koff + 1];
#pragma unroll
    for (int tt = 0; tt < 4; ++tt) {
      v2f b;
      const float* bcol = W2 + tt * 16 + lo;
      b.x = bcol[(size_t)(kk + koff) * NCLS];
      b.y = bcol[(size_t)(kk + koff + 1) * NCLS];
      acc[tt] = __builtin_amdgcn_wmma_f32_16x16x4_f32(false, a, false, b, (short)0, acc[tt],
                                                      false, false);
    }
  }
  // bias
#pragma unroll
  for (int tt = 0; tt < 4; ++tt) {
    float bias = b2[tt * 16 + lo];
#pragma unroll
    for (int v = 0; v < 8; ++v) acc[tt][v] += bias;
  }
  const int hi = lane >> 4;
  // softmax per row: each row lives in 16 lanes (one half) x 4 regs
#pragma unroll
  for (int v = 0; v < 8; ++v) {
    float m = fmaxf(fmaxf(acc[0][v], acc[1][v]), fmaxf(acc[2][v], acc[3][v]));
#pragma unroll
    for (int d = 8; d >= 1; d >>= 1) m = fmaxf(m, __shfl_xor(m, d, 32));
    float e0 = expf(acc[0][v] - m), e1 = expf(acc[1][v] - m);
    float e2 = expf(acc[2][v] - m), e3 = expf(acc[3][v] - m);
    float s = e0 + e1 + e2 + e3;
#pragma unroll
    for (int d = 8; d >= 1; d >>= 1) s += __shfl_xor(s, d, 32);
    float inv = 1.0f / s;
    float* op = out + (size_t)(row0 + v + hi * 8) * NCLS + lo;
    op[0]  = e0 * inv;
    op[16] = e1 * inv;
    op[32] = e2 * inv;
    op[48] = e3 * inv;
  }
}

// ---------------------------------------------------------------------------
// Host launcher
// ---------------------------------------------------------------------------
extern "C" void kernel_launch(void* const* d_in, const int* in_sizes, int n_in,
                              void* d_out, int out_size, void* d_ws, size_t ws_size,
                              hipStream_t stream) {
  const float* x      = (const float*)d_in[0];
  const int*   src    = (const int*)d_in[1];             // edge_index[0]
  const int*   dst    = (const int*)d_in[1] + NEDGES;    // edge_index[1]
  const float* Wg     = (const float*)d_in[2];
  const float* bg     = (const float*)d_in[3];
  const float* ln_g   = (const float*)d_in[4];
  const float* ln_b   = (const float*)d_in[5];
  const float* W1     = (const float*)d_in[6];
  const float* b1     = (const float*)d_in[7];
  const float* W2     = (const float*)d_in[8];
  const float* b2     = (const float*)d_in[9];
  float* out = (float*)d_out;

  // Workspace layout (floats): dinv | xc | hW (reused as z1) | agg
  float* dinv = (float*)d_ws;
  float* xc   = dinv + NNODES;
  float* hW   = xc + (size_t)NNODES * XCDIM;
  float* agg  = hW + (size_t)NNODES * HDIM;
  float* z1   = hW;  // alias: hW dead after layer loop

  // degrees -> dinv
  fill_ones_kernel<<<(NNODES + 255) / 256, 256, 0, stream>>>(dinv, NNODES);
  deg_accum_kernel<<<(NEDGES + 255) / 256, 256, 0, stream>>>(dst, dinv, NEDGES);
  rsqrt_inplace_kernel<<<(NNODES + 255) / 256, 256, 0, stream>>>(dinv, NNODES);

  // xc[:, :128] = x
  copy_x_kernel<<<(NNODES * 32 + 255) / 256, 256, 0, stream>>>(x, xc);

  for (int i = 0; i < NLAYERS; ++i) {
    const float* h   = (i == 0) ? x : (xc + i * HDIM);
    const int    lda = (i == 0) ? DFEAT : XCDIM;
    gemm128_kernel<<<ROWTILES, 256, 0, stream>>>(h, lda, Wg + (size_t)i * DFEAT * HDIM, hW);
    selfloop_kernel<<<(NNODES * 32 + 255) / 256, 256, 0, stream>>>(hW, dinv, agg);
    scatter_kernel<<<(NEDGES * 32 + 255) / 256, 256, 0, stream>>>(src, dst, dinv, hW, agg);
    post_ln_kernel<<<(NNODES * 32 + 255) / 256, 256, 0, stream>>>(
        agg, bg + i * HDIM, ln_g + i * HDIM, ln_b + i * HDIM, xc, (i + 1) * HDIM);
  }

  mlp1_kernel<<<ROWTILES, 256, 0, stream>>>(xc, W1, b1, z1);
  mlp2_softmax_kernel<<<(ROWTILES * 32 + 127) / 128, 128, 0, stream>>>(z1, W2, b2, out);
}